// ODEfunc_72335839199608
// MI455X (gfx1250) — compile-verified
//
#include <hip/hip_runtime.h>

typedef float v2f __attribute__((ext_vector_type(2)));
typedef float v8f __attribute__((ext_vector_type(8)));

#define HIDDEN 128
#define LDSW   132   // padded LDS pitch: lanes hit distinct banks on B reads

__device__ __forceinline__ float sigmoidf(float t) {
    return 1.0f / (1.0f + __expf(-t));
}

// ---------------------------------------------------------------------------
// Kernel 1: S/I = sigmoid(x[s] @ W^T + b) for s in {0,1}, via f32 WMMA.
// One wave computes a 16-node x 128-col output strip. W cached in LDS.
// ---------------------------------------------------------------------------
__global__ __launch_bounds__(256) void gemm_sigmoid_kernel(
    const float* __restrict__ x, const float* __restrict__ W,
    const float* __restrict__ bias, float* __restrict__ out,
    int n_nodes, int n_tiles)
{
    __shared__ float Wlds[HIDDEN * LDSW];
    const int t = threadIdx.x;

    // Cooperative load of W (128x128 f32 = 64KB) into LDS, pitch 132.
#pragma unroll
    for (int i = 0; i < 16; ++i) {
        const int flat4 = t + 256 * i;      // 0..4095 float4s
        const int row   = flat4 >> 5;       // 32 float4 per row
        const int c4    = flat4 & 31;
        const float4 v  = ((const float4*)W)[flat4];
        float* dst = &Wlds[row * LDSW + c4 * 4];
        dst[0] = v.x; dst[1] = v.y; dst[2] = v.z; dst[3] = v.w;
    }
    __syncthreads();

    const int wave = t >> 5;
    const int lane = t & 31;
    const int wg   = blockIdx.x * 8 + wave;
    if (wg >= 2 * n_tiles) return;

    const int s    = wg / n_tiles;
    const int tile = wg - s * n_tiles;
    const int row0 = tile * 16;
    const int half = lane >> 4;    // 0: K={0,1} of each K4-chunk, 1: K={2,3}
    const int lm   = lane & 15;    // A: row-in-tile, B/D: column-in-ntile

    const float* arow = x + ((size_t)s * n_nodes + (size_t)(row0 + lm)) * HIDDEN;

    v8f acc[8] = {};

    for (int kb = 0; kb < 32; ++kb) {
        const int k0 = kb * 4 + half * 2;
        const v2f a = *(const v2f*)(arow + k0);         // A[m][k0..k0+1]
#pragma unroll
        for (int nt = 0; nt < 8; ++nt) {
            const int n = nt * 16 + lm;
            const v2f b = *(const v2f*)(&Wlds[n * LDSW + k0]); // B[k][n]=W[n][k]
            acc[nt] = __builtin_amdgcn_wmma_f32_16x16x4_f32(
                false, a, false, b, (short)0, acc[nt], false, false);
        }
    }

    float* obase = out + (size_t)s * n_nodes * HIDDEN;
#pragma unroll
    for (int nt = 0; nt < 8; ++nt) {
        const int n  = nt * 16 + lm;
        const float bn = bias[n];
#pragma unroll
        for (int r = 0; r < 8; ++r) {
            const int m = row0 + r + 8 * half;
            if (m < n_nodes)
                obase[(size_t)m * HIDDEN + n] = sigmoidf(acc[nt][r] + bn);
        }
    }
}

// ---------------------------------------------------------------------------
// Kernel 0: zero the AI accumulator workspace.
// ---------------------------------------------------------------------------
__global__ __launch_bounds__(256) void zero_kernel(float4* __restrict__ p, int n4) {
    const int i = blockIdx.x * blockDim.x + threadIdx.x;
    if (i < n4) p[i] = make_float4(0.f, 0.f, 0.f, 0.f);
}

// ---------------------------------------------------------------------------
// Kernel 2: AI[rows[e]] += I[cols[e]]. One wave per edge; lane handles 4
// floats: float4 gather + 4 HW f32 atomics (L2-resident, 51MB << 192MB L2).
// ---------------------------------------------------------------------------
__global__ __launch_bounds__(256) void edge_scatter_kernel(
    const int* __restrict__ rows, const int* __restrict__ cols,
    const float* __restrict__ I, float* __restrict__ AI, int n_edges)
{
    const int wave = threadIdx.x >> 5;
    const int lane = threadIdx.x & 31;
    const int e = blockIdx.x * 8 + wave;
    if (e >= n_edges) return;

    const int r = rows[e];
    const int c = cols[e];
    const float4 vi = *(const float4*)(I + (size_t)c * HIDDEN + lane * 4);
    float* dst = AI + (size_t)r * HIDDEN + lane * 4;
    unsafeAtomicAdd(dst + 0, vi.x);
    unsafeAtomicAdd(dst + 1, vi.y);
    unsafeAtomicAdd(dst + 2, vi.z);
    unsafeAtomicAdd(dst + 3, vi.w);
}

// ---------------------------------------------------------------------------
// Kernel 3: SIR dynamics. S/I live in out slabs 0/1 and are overwritten
// in place by dS/dI; dR -> slab 2; zeros -> slab 3.
// ---------------------------------------------------------------------------
__global__ __launch_bounds__(256) void finalize_kernel(
    const float* __restrict__ x3, const float* __restrict__ AI,
    float* __restrict__ out, int n_nodes)
{
    const size_t NH  = (size_t)n_nodes * HIDDEN;
    const size_t n4t = NH >> 2;
    const size_t i4  = (size_t)blockIdx.x * blockDim.x + threadIdx.x;
    if (i4 >= n4t) return;

    const size_t i   = i4 * 4;
    const int node   = (int)(i >> 7);
    const float beta  = x3[(size_t)node * HIDDEN + 0];
    const float gamma = x3[(size_t)node * HIDDEN + 1];

    float4 S4 = ((const float4*)out)[i4];
    float4 I4 = ((const float4*)(out + NH))[i4];
    float4 A4 = ((const float4*)AI)[i4];
    float4 o0, o1, o2, o3;

    const float* Sp = (const float*)&S4;
    const float* Ip = (const float*)&I4;
    const float* Ap = (const float*)&A4;
    float* p0 = (float*)&o0; float* p1 = (float*)&o1;
    float* p2 = (float*)&o2; float* p3 = (float*)&o3;
#pragma unroll
    for (int j = 0; j < 4; ++j) {
        const float dS = -beta * Ap[j] * Sp[j];
        p0[j] = dS;
        p1[j] = -dS - gamma * Ip[j];
        p2[j] = gamma * Ip[j];
        p3[j] = 0.0f;
    }
    ((float4*)out)[i4]            = o0;
    ((float4*)(out + NH))[i4]     = o1;
    ((float4*)(out + 2 * NH))[i4] = o2;
    ((float4*)(out + 3 * NH))[i4] = o3;
}

// ---------------------------------------------------------------------------
extern "C" void kernel_launch(void* const* d_in, const int* in_sizes, int n_in,
                              void* d_out, int out_size, void* d_ws, size_t ws_size,
                              hipStream_t stream) {
    const float* x    = (const float*)d_in[0];
    const float* W    = (const float*)d_in[1];
    const float* b    = (const float*)d_in[2];
    const int*   rows = (const int*)d_in[3];
    const int*   cols = (const int*)d_in[4];
    float* out = (float*)d_out;
    float* AI  = (float*)d_ws;

    const int n_nodes = in_sizes[0] / (4 * HIDDEN);
    const int n_edges = in_sizes[3];
    const int n_tiles = (n_nodes + 15) / 16;

    // AI := 0
    const int n4 = (n_nodes * HIDDEN) / 4;
    zero_kernel<<<(n4 + 255) / 256, 256, 0, stream>>>((float4*)AI, n4);

    // S -> out slab 0, I -> out slab 1
    const int gemm_blocks = (2 * n_tiles + 7) / 8;
    gemm_sigmoid_kernel<<<gemm_blocks, 256, 0, stream>>>(x, W, b, out, n_nodes, n_tiles);

    // AI[rows] += I[cols]
    edge_scatter_kernel<<<(n_edges + 7) / 8, 256, 0, stream>>>(
        rows, cols, out + (size_t)n_nodes * HIDDEN, AI, n_edges);

    // dynamics + write all four output slabs
    const size_t fin_threads = ((size_t)n_nodes * HIDDEN) / 4;
    finalize_kernel<<<(int)((fin_threads + 255) / 256), 256, 0, stream>>>(
        x + (size_t)3 * n_nodes * HIDDEN, AI, out, n_nodes);
}